// SnnModel_90761248899469
// MI455X (gfx1250) — compile-verified
//
#include <hip/hip_runtime.h>

#define B_ROWS 262144
#define D_IN 128
#define L1 128
#define L2 64
#define D_OUT 2
#define T_STEPS 5
#define THR 1.0f
#define BETA_LIF 0.9f
#define BN_EPS 1e-5f
#define W3_STRIDE 132  // padded LDS stride: banks 4*col+k all distinct across wave

typedef float v2f __attribute__((ext_vector_type(2)));
typedef float v8f __attribute__((ext_vector_type(8)));

__device__ __forceinline__ v8f wmma4(v2f a, v2f b, v8f c) {
  // V_WMMA_F32_16X16X4_F32 : D = A(16x4) * B(4x16) + C(16x16), full f32
  return __builtin_amdgcn_wmma_f32_16x16x4_f32(
      /*neg_a=*/false, a, /*neg_b=*/false, b,
      /*c_mod=*/(short)0, c, /*reuse_a=*/false, /*reuse_b=*/false);
}

// ---------------------------------------------------------------------------
// Kernel 1: h1 = x @ W1^T + b1  (f32 WMMA), plus BN1 batch statistics.
// One wave = 16 rows x 128 cols. A fragments (x) live in 64 VGPRs.
// ---------------------------------------------------------------------------
__global__ __launch_bounds__(256) void k_gemm1(
    const float* __restrict__ x, const float* __restrict__ W1,
    const float* __restrict__ b1, float* __restrict__ h1,
    float* __restrict__ stat /* [0..127]=sum, [128..255]=sumsq */) {
  __shared__ float s_sum[L1];
  __shared__ float s_sq[L1];
  const int tid = threadIdx.x;
  for (int i = tid; i < L1; i += 256) { s_sum[i] = 0.f; s_sq[i] = 0.f; }
  __syncthreads();

  const int lane = tid & 31, wave = tid >> 5;
  const int hi = lane >> 4, col = lane & 15;
  const int r0 = (blockIdx.x * 8 + wave) * 16;
  const int row = r0 + col;

  v2f a[32];
#pragma unroll
  for (int t = 0; t < 32; ++t) {
    const int k = 4 * t + 2 * hi;
    a[t] = *(const v2f*)(x + (size_t)row * D_IN + k);
  }

  for (int n = 0; n < 8; ++n) {
    v8f c = {0.f, 0.f, 0.f, 0.f, 0.f, 0.f, 0.f, 0.f};
    const float* wb = W1 + (size_t)(n * 16 + col) * D_IN;
#pragma unroll
    for (int t = 0; t < 32; ++t) {
      const int k = 4 * t + 2 * hi;
      v2f b = *(const v2f*)(wb + k);
      c = wmma4(a[t], b, c);
    }
    const int j = n * 16 + col;
    const float bias = b1[j];
    float s = 0.f, q = 0.f;
#pragma unroll
    for (int r = 0; r < 8; ++r) {
      float v = c[r] + bias;
      h1[(size_t)(r0 + r + 8 * hi) * L1 + j] = v;
      s += v;
      q += v * v;
    }
    atomicAdd(&s_sum[j], s);
    atomicAdd(&s_sq[j], q);
  }
  __syncthreads();
  for (int i = tid; i < L1; i += 256) {
    atomicAdd(stat + i, s_sum[i]);
    atomicAdd(stat + L1 + i, s_sq[i]);
  }
}

// ---------------------------------------------------------------------------
// BN1: fold batch statistics into per-feature scale/shift.
// ---------------------------------------------------------------------------
__global__ void k_bnfin(const float* __restrict__ stat,
                        const float* __restrict__ g, const float* __restrict__ bb,
                        float* __restrict__ scale, float* __restrict__ shift,
                        int n) {
  const int i = blockIdx.x * blockDim.x + threadIdx.x;
  if (i < n) {
    const float inv = 1.0f / (float)B_ROWS;
    const float m = stat[i] * inv;
    const float var = stat[n + i] * inv - m * m;
    const float rs = rsqrtf(var + BN_EPS);
    const float sc = g[i] * rs;
    scale[i] = sc;
    shift[i] = bb[i] - m * sc;
  }
}

// BN3 for all 5 steps in one launch: blockIdx.x = time step.
__global__ void k_bnfin3(const float* __restrict__ stat3,
                         const float* __restrict__ g, const float* __restrict__ bb,
                         float* __restrict__ scale3, float* __restrict__ shift3) {
  const int t = blockIdx.x;
  const int i = threadIdx.x;  // 64 threads
  const float* stat = stat3 + t * 2 * L2;
  const float inv = 1.0f / (float)B_ROWS;
  const float m = stat[i] * inv;
  const float var = stat[L2 + i] * inv - m * m;
  const float rs = rsqrtf(var + BN_EPS);
  const float sc = g[i] * rs;
  scale3[t * L2 + i] = sc;
  shift3[t * L2 + i] = bb[i] - m * sc;
}

// ---------------------------------------------------------------------------
// Fused layer-1 recurrence + GEMM2 over ALL 5 time steps.
// cur1 and mem1 never touch memory: both live in registers (64+64 VGPRs).
// Spikes are recomputed from mem1 inside the WMMA loop (co-executes with XDL).
// W3 staged once in LDS (padded stride 132 -> conflict-free ds_load_b64).
// ---------------------------------------------------------------------------
__global__ __launch_bounds__(256) void k_lif1_gemm2(
    const float* __restrict__ h1, const float* __restrict__ scale1,
    const float* __restrict__ shift1,
    const float* __restrict__ W3, const float* __restrict__ b3,
    float* __restrict__ cur3, float* __restrict__ stat3) {
  __shared__ float s_w3[L2 * W3_STRIDE];
  __shared__ float s_sum[L2];
  __shared__ float s_sq[L2];
  const int tid = threadIdx.x;
  for (int i = tid; i < L2 * 128; i += 256) {
    const int r = i >> 7, c = i & 127;
    s_w3[r * W3_STRIDE + c] = W3[i];
  }

  const int lane = tid & 31, wave = tid >> 5;
  const int hi = lane >> 4, col = lane & 15;
  const int r0 = (blockIdx.x * 8 + wave) * 16;
  const int row = r0 + col;

  const v2f zero = {0.f, 0.f};
  v2f cur1[32], mem[32];
#pragma unroll
  for (int t = 0; t < 32; ++t) {
    const int k = 4 * t + 2 * hi;
    v2f h = *(const v2f*)(h1 + (size_t)row * L1 + k);
    v2f sc = *(const v2f*)(scale1 + k);
    v2f sh = *(const v2f*)(shift1 + k);
    cur1[t] = h * sc + sh;  // BN1 applied once, held in registers
    mem[t] = zero;
  }
  float bias[4];
#pragma unroll
  for (int n = 0; n < 4; ++n) bias[n] = b3[n * 16 + col];

  __syncthreads();  // W3 staged

  for (int st = 0; st < T_STEPS; ++st) {
    if (tid < L2) { s_sum[tid] = 0.f; s_sq[tid] = 0.f; }
    __syncthreads();

    // LIF1: reset on pre-update membrane, leaky integrate, stay in registers.
#pragma unroll
    for (int t = 0; t < 32; ++t) {
      v2f m = mem[t];
      v2f mn;
      mn.x = BETA_LIF * m.x + cur1[t].x - ((m.x > THR) ? THR : 0.f);
      mn.y = BETA_LIF * m.y + cur1[t].y - ((m.y > THR) ? THR : 0.f);
      mem[t] = mn;
    }

    float* curT = cur3 + (size_t)st * B_ROWS * L2;
    float* statT = stat3 + st * 2 * L2;
    for (int n = 0; n < 4; ++n) {
      v8f c = {0.f, 0.f, 0.f, 0.f, 0.f, 0.f, 0.f, 0.f};
      const float* wb = s_w3 + (n * 16 + col) * W3_STRIDE;
#pragma unroll
      for (int t = 0; t < 32; ++t) {
        const int k = 4 * t + 2 * hi;
        v2f a;  // spike recomputed on the fly: keeps VGPR pressure down
        a.x = (mem[t].x > THR) ? 1.f : 0.f;
        a.y = (mem[t].y > THR) ? 1.f : 0.f;
        v2f b = *(const v2f*)(wb + k);
        c = wmma4(a, b, c);
      }
      const int j = n * 16 + col;
      float s = 0.f, q = 0.f;
#pragma unroll
      for (int r = 0; r < 8; ++r) {
        float v = c[r] + bias[n];
        curT[(size_t)(r0 + r + 8 * hi) * L2 + j] = v;
        s += v;
        q += v * v;
      }
      atomicAdd(&s_sum[j], s);
      atomicAdd(&s_sq[j], q);
    }
    __syncthreads();
    if (tid < L2) {
      atomicAdd(statT + tid, s_sum[tid]);
      atomicAdd(statT + L2 + tid, s_sq[tid]);
    }
    __syncthreads();
  }
}

// ---------------------------------------------------------------------------
// Fused tail over ALL 5 steps: BN3 apply + LIF3 + (spk3 @ W4^T + b4) + LIF4.
// mem3 (4 regs/lane) and mem4 (2 regs) live in registers across steps.
// One 16-lane half-wave per row; K=64 dot via shfl_xor butterfly.
// ---------------------------------------------------------------------------
__global__ __launch_bounds__(256) void k_tail(
    const float* __restrict__ cur3, const float* __restrict__ scale3,
    const float* __restrict__ shift3, const float* __restrict__ W4,
    const float* __restrict__ b4, const float* __restrict__ beta4,
    float* __restrict__ out) {
  const int tid = threadIdx.x;
  const int row = blockIdx.x * 16 + (tid >> 4);
  const int j0 = tid & 15;

  float w40[4], w41[4], m3[4];
#pragma unroll
  for (int s = 0; s < 4; ++s) {
    w40[s] = W4[j0 + 16 * s];
    w41[s] = W4[L2 + j0 + 16 * s];
    m3[s] = 0.f;
  }
  const float bc0 = fminf(fmaxf(beta4[0], 0.f), 1.f);
  const float bc1 = fminf(fmaxf(beta4[1], 0.f), 1.f);
  const float bias0 = b4[0], bias1 = b4[1];
  float m40 = 0.f, m41 = 0.f;

  float* out_spk = out;
  float* out_mem = out + (size_t)T_STEPS * B_ROWS * D_OUT;

  for (int t = 0; t < T_STEPS; ++t) {
    const float* curT = cur3 + (size_t)t * B_ROWS * L2;
    const float* scT = scale3 + t * L2;
    const float* shT = shift3 + t * L2;
    float p0 = 0.f, p1 = 0.f;
#pragma unroll
    for (int s = 0; s < 4; ++s) {
      const int j = j0 + 16 * s;
      const float v = curT[(size_t)row * L2 + j] * scT[j] + shT[j];
      const float m = m3[s];
      const float mn = BETA_LIF * m + v - ((m > THR) ? THR : 0.f);
      m3[s] = mn;
      const float spk = (mn > THR) ? 1.f : 0.f;
      p0 += spk * w40[s];
      p1 += spk * w41[s];
    }
#pragma unroll
    for (int off = 1; off < 16; off <<= 1) {
      p0 += __shfl_xor(p0, off, 32);
      p1 += __shfl_xor(p1, off, 32);
    }
    // butterfly leaves the full sum in every lane of the half-wave
    m40 = bc0 * m40 + p0 + bias0;
    m41 = bc1 * m41 + p1 + bias1;
    if (j0 == 0) {
      const size_t o = (size_t)t * B_ROWS * D_OUT + (size_t)row * 2;
      out_spk[o + 0] = (m40 > THR) ? 1.f : 0.f;
      out_spk[o + 1] = (m41 > THR) ? 1.f : 0.f;
      out_mem[o + 0] = m40;
      out_mem[o + 1] = m41;
    }
  }
}

extern "C" void kernel_launch(void* const* d_in, const int* in_sizes, int n_in,
                              void* d_out, int out_size, void* d_ws, size_t ws_size,
                              hipStream_t stream) {
  (void)in_sizes; (void)n_in; (void)out_size; (void)ws_size;
  const float* x     = (const float*)d_in[0];
  const float* W1    = (const float*)d_in[1];
  const float* b1    = (const float*)d_in[2];
  const float* g1    = (const float*)d_in[3];
  const float* bb1   = (const float*)d_in[4];
  const float* W3    = (const float*)d_in[5];
  const float* b3    = (const float*)d_in[6];
  const float* g2    = (const float*)d_in[7];
  const float* bb2   = (const float*)d_in[8];
  const float* W4    = (const float*)d_in[9];
  const float* b4    = (const float*)d_in[10];
  const float* beta4 = (const float*)d_in[11];
  float* out = (float*)d_out;

  float* ws = (float*)d_ws;
  size_t off = 0;
  float* h1     = ws + off; off += (size_t)B_ROWS * L1;            // 134 MB
  float* cur3   = ws + off; off += (size_t)T_STEPS * B_ROWS * L2;  // 335 MB
  float* stat1  = ws + off; off += 2 * L1;   // |-- small tail, zeroed as one block
  float* scale1 = ws + off; off += L1;
  float* shift1 = ws + off; off += L1;
  float* stat3  = ws + off; off += T_STEPS * 2 * L2;
  float* scale3 = ws + off; off += T_STEPS * L2;
  float* shift3 = ws + off; off += T_STEPS * L2;
  const size_t tail_floats = off - ((size_t)B_ROWS * L1 + (size_t)T_STEPS * B_ROWS * L2);

  // Only the tiny stat/scale/shift tail needs deterministic zeroing.
  hipMemsetAsync(stat1, 0, sizeof(float) * tail_floats, stream);

  const dim3 blk(256);
  k_gemm1<<<B_ROWS / 128, blk, 0, stream>>>(x, W1, b1, h1, stat1);
  k_bnfin<<<1, 128, 0, stream>>>(stat1, g1, bb1, scale1, shift1, L1);
  k_lif1_gemm2<<<B_ROWS / 128, blk, 0, stream>>>(h1, scale1, shift1, W3, b3,
                                                 cur3, stat3);
  k_bnfin3<<<T_STEPS, L2, 0, stream>>>(stat3, g2, bb2, scale3, shift3);
  k_tail<<<B_ROWS / 16, blk, 0, stream>>>(cur3, scale3, shift3, W4, b4, beta4, out);
}